// FeaStConv_83330955477200
// MI455X (gfx1250) — compile-verified
//
#include <hip/hip_runtime.h>
#include <hip/hip_bf16.h>

// FeaStConv, restructured:
//   y = x @ W           (N x 256)  -- WMMA f32 16x16x4 GEMM, fused with
//   s = x @ U           (N x 2)
//   per edge e: q = softmax(s[src]-s[dst]+c); atomically agg[dst] += q0*y[src,0:128]+q1*y[src,128:256]
//   out = x + relu(agg/max(deg,1) + bias)

#define D_FEAT 128
#define HD     256   // H * D

typedef float v2f __attribute__((ext_vector_type(2)));
typedef float v8f __attribute__((ext_vector_type(8)));

// ---------------------------------------------------------------------------
// Kernel 0: zero the accumulator (d_out) and degree buffer
// ---------------------------------------------------------------------------
__global__ __launch_bounds__(256)
void feast_zero_kernel(float* __restrict__ agg, float* __restrict__ deg,
                       int nAgg, int nDeg) {
    const int stride = gridDim.x * blockDim.x;
    for (int j = blockIdx.x * blockDim.x + threadIdx.x; j < nAgg; j += stride)
        agg[j] = 0.0f;
    for (int j = blockIdx.x * blockDim.x + threadIdx.x; j < nDeg; j += stride)
        deg[j] = 0.0f;
}

// ---------------------------------------------------------------------------
// Kernel 1: y = x @ W  (M x 128 @ 128 x 256) via V_WMMA_F32_16X16X4_F32,
//           fused s = x @ U (M x 2).
// One wave per 16-row strip; wave iterates K in steps of 4 and owns all 16
// column tiles (acc = 16 * v8f = 128 VGPRs), so x is streamed once from HBM.
//
// f32 WMMA VGPR layouts (ISA 7.12.2):
//   A 16x4 : lane l holds row m=l&15, ks {kk, kk+1}, kk = 2*(l>>4)
//   B 4x16 : lane l holds col n=l&15, ks {kk, kk+1}
//   C 16x16: lane l, vgpr v -> row v + 8*(l>>4), col l&15
// ---------------------------------------------------------------------------
__global__ __launch_bounds__(256)
void feast_gemm_kernel(const float* __restrict__ x, const float* __restrict__ W,
                       const float* __restrict__ U, float* __restrict__ y,
                       float* __restrict__ s, int nrows) {
    const int lane  = threadIdx.x & 31;
    const int wave  = threadIdx.x >> 5;
    const int mtiles = (nrows + 15) >> 4;
    const int mtile  = blockIdx.x * 8 + wave;
    if (mtile >= mtiles) return;

    const int row0  = mtile * 16;
    const int mrow  = lane & 15;   // A row / C col index
    const int khalf = lane >> 4;   // which K pair this lane holds
    const int ncol  = lane & 15;   // B col

    const float* __restrict__ Arow = x + (size_t)(row0 + mrow) * D_FEAT;

    v8f acc[16] = {};              // 16 column tiles of the 16x256 strip
    float su0 = 0.0f, su1 = 0.0f;  // s = x @ U partials (this lane's K slices)

    for (int k = 0; k < 32; ++k) {
        const int kk = 4 * k + 2 * khalf;

        v2f a;
        a.x = Arow[kk];
        a.y = Arow[kk + 1];

        // U row-major [128][2]: the 4 floats U[kk][0..1],U[kk+1][0..1] are
        // contiguous and 16B aligned (kk even).
        const float4 u4 = *reinterpret_cast<const float4*>(U + 2 * kk);
        su0 = fmaf(a.x, u4.x, su0);
        su1 = fmaf(a.x, u4.y, su1);
        su0 = fmaf(a.y, u4.z, su0);
        su1 = fmaf(a.y, u4.w, su1);

        const float* __restrict__ Wk0 = W + (size_t)kk * HD + ncol;
        const float* __restrict__ Wk1 = Wk0 + HD;
#pragma unroll
        for (int nt = 0; nt < 16; ++nt) {
            v2f b;
            b.x = Wk0[nt * 16];
            b.y = Wk1[nt * 16];
            // 8 args: (neg_a, A, neg_b, B, c_mod, C, reuse_a, reuse_b)
            acc[nt] = __builtin_amdgcn_wmma_f32_16x16x4_f32(
                false, a, false, b, (short)0, acc[nt], false, false);
        }
    }

    // Combine K-halves (lanes l and l^16 together cover K=0..127 of row mrow)
    su0 += __shfl_xor(su0, 16, 32);
    su1 += __shfl_xor(su1, 16, 32);
    if (khalf == 0) {
        float2* sp = reinterpret_cast<float2*>(s);
        sp[row0 + mrow] = make_float2(su0, su1);
    }

    // Store D tiles
#pragma unroll
    for (int nt = 0; nt < 16; ++nt) {
#pragma unroll
        for (int v = 0; v < 8; ++v) {
            y[(size_t)(row0 + v + 8 * khalf) * HD + nt * 16 + ncol] = acc[nt][v];
        }
    }
}

// ---------------------------------------------------------------------------
// Kernel 2: edge scatter. One wave per edge (E real edges + N self loops).
// All 32 lanes uniformly compute the 2-way softmax; lanes gather y[src] as
// float4 (1KB/edge, coalesced) and scatter with hardware fp32 atomics.
// ---------------------------------------------------------------------------
__global__ __launch_bounds__(256)
void feast_edge_kernel(const float* __restrict__ y, const float2* __restrict__ s,
                       const long long* __restrict__ ei,
                       const float* __restrict__ cvec,
                       float* __restrict__ agg, float* __restrict__ deg,
                       int E, int Nn) {
    const int lane = threadIdx.x & 31;
    const long long total = (long long)E + (long long)Nn;
    const int wid = (blockIdx.x * blockDim.x + threadIdx.x) >> 5;
    const int nw  = (gridDim.x * blockDim.x) >> 5;
    const float c0 = cvec[0];
    const float c1 = cvec[1];

    for (long long e = wid; e < total; e += nw) {
        int srcN, dstN;
        if (e < (long long)E) {
            srcN = (int)ei[e];
            dstN = (int)ei[(long long)E + e];
        } else {
            srcN = dstN = (int)(e - (long long)E);   // self loop
        }

        const float2 ss = s[srcN];
        const float2 sd = s[dstN];
        float t0 = ss.x - sd.x + c0;
        float t1 = ss.y - sd.y + c1;
        const float mx = fmaxf(t0, t1);
        const float e0 = __expf(t0 - mx);
        const float e1 = __expf(t1 - mx);
        const float inv = 1.0f / (e0 + e1);
        const float q0 = e0 * inv;
        const float q1 = e1 * inv;

        const float4* __restrict__ yr =
            reinterpret_cast<const float4*>(y + (size_t)srcN * HD);
        const float4 a0 = yr[lane];        // head 0, feats lane*4..lane*4+3
        const float4 a1 = yr[32 + lane];   // head 1

        float* ap = agg + (size_t)dstN * D_FEAT + lane * 4;
        unsafeAtomicAdd(ap + 0, q0 * a0.x + q1 * a1.x);
        unsafeAtomicAdd(ap + 1, q0 * a0.y + q1 * a1.y);
        unsafeAtomicAdd(ap + 2, q0 * a0.z + q1 * a1.z);
        unsafeAtomicAdd(ap + 3, q0 * a0.w + q1 * a1.w);
        if (lane == 0) unsafeAtomicAdd(deg + dstN, 1.0f);
    }
}

// ---------------------------------------------------------------------------
// Kernel 3: out = x + relu(agg/max(deg,1) + bias), in place on d_out.
// ---------------------------------------------------------------------------
__global__ __launch_bounds__(256)
void feast_finalize_kernel(const float* __restrict__ x,
                           const float* __restrict__ bias,
                           const float* __restrict__ deg,
                           float* __restrict__ out, int Nn) {
    const int idx = blockIdx.x * blockDim.x + threadIdx.x; // one per 4 feats
    const int total = Nn * (D_FEAT / 4);
    if (idx >= total) return;
    const int node = idx >> 5;
    const int grp  = idx & 31;

    float4* o4 = reinterpret_cast<float4*>(out);
    const float4* x4 = reinterpret_cast<const float4*>(x);
    const float4* b4 = reinterpret_cast<const float4*>(bias);

    const float invd = 1.0f / fmaxf(deg[node], 1.0f);
    const float4 a  = o4[idx];
    const float4 xb = x4[idx];
    const float4 bb = b4[grp];

    float4 r;
    r.x = xb.x + fmaxf(fmaf(a.x, invd, bb.x), 0.0f);
    r.y = xb.y + fmaxf(fmaf(a.y, invd, bb.y), 0.0f);
    r.z = xb.z + fmaxf(fmaf(a.z, invd, bb.z), 0.0f);
    r.w = xb.w + fmaxf(fmaf(a.w, invd, bb.w), 0.0f);
    o4[idx] = r;
}

// ---------------------------------------------------------------------------
extern "C" void kernel_launch(void* const* d_in, const int* in_sizes, int n_in,
                              void* d_out, int out_size, void* d_ws, size_t ws_size,
                              hipStream_t stream) {
    const float*     x    = (const float*)d_in[0];
    const long long* ei   = (const long long*)d_in[1];
    const float*     W    = (const float*)d_in[2];
    const float*     U    = (const float*)d_in[3];
    const float*     c    = (const float*)d_in[4];
    const float*     bias = (const float*)d_in[5];

    const int Nn = in_sizes[0] / D_FEAT;   // 100000
    const int E  = in_sizes[1] / 2;        // 500000

    float* out = (float*)d_out;            // also used as the atomic accumulator

    float* y   = (float*)d_ws;             // Nn * 256 f32
    float* s   = y + (size_t)Nn * HD;      // Nn * 2 f32
    float* deg = s + (size_t)Nn * 2;       // Nn f32

    // 0) zero accumulator + degrees
    feast_zero_kernel<<<2048, 256, 0, stream>>>(out, deg, Nn * D_FEAT, Nn);

    // 1) y = x@W (WMMA f32), s = x@U (fused)
    const int mtiles = (Nn + 15) / 16;
    feast_gemm_kernel<<<(mtiles + 7) / 8, 256, 0, stream>>>(x, W, U, y, s, Nn);

    // 2) per-edge softmax-weighted scatter (1 wave per edge)
    const long long ET = (long long)E + Nn;
    int eblocks = (int)((ET + 7) / 8);     // 8 waves per 256-thread block
    feast_edge_kernel<<<eblocks, 256, 0, stream>>>(
        y, (const float2*)s, ei, c, out, deg, E, Nn);

    // 3) mean + bias + relu + residual
    const int ftotal = Nn * (D_FEAT / 4);
    feast_finalize_kernel<<<(ftotal + 255) / 256, 256, 0, stream>>>(
        x, bias, deg, out, Nn);
}